// Scaled_Dot_Product_Attention_32315333935614
// MI455X (gfx1250) — compile-verified
//
#include <hip/hip_runtime.h>
#include <math.h>

typedef __attribute__((ext_vector_type(2))) float v2f;
typedef __attribute__((ext_vector_type(4))) float v4f;
typedef __attribute__((ext_vector_type(8))) float v8f;

#define B_   2
#define H_   16
#define S_   2048
#define D_   64
#define SROW 2052   // 2048 + 4 floats pad -> rows skew across the 64 LDS banks

// One block = 16 query rows of one (b,h). 8 waves, each wave owns every 8th
// 16-key column tile. Raw masked scores are staged in LDS (16 x 2052 f32),
// then softmax'd in-place and streamed out with nontemporal b128 stores.
// Streaming traffic (mask in, score out) is marked non-temporal so the tiny,
// heavily reused Q/K working set stays resident in WGP$/L2.
__global__ __launch_bounds__(256, 2)
void attn_score_kernel(const float* __restrict__ q,
                       const float* __restrict__ kmat,
                       const int*   __restrict__ mask,
                       float*       __restrict__ out)
{
    extern __shared__ float sc[];          // 16 * SROW floats (dynamic)
    __shared__ float pm[16][17];
    __shared__ float ps[16][17];
    __shared__ float rowmax[16];
    __shared__ float rowinv[16];

    const int t    = threadIdx.x;
    const int lane = t & 31;
    const int wave = t >> 5;

    const int blk = blockIdx.x;
    const int bh  = blk >> 7;              // 0..31  (b*H + h)
    const int q0  = (blk & 127) << 4;      // first query row of this tile

    const int m  = lane & 15;              // row (A) / col (B,C) index
    const int hi = lane >> 4;
    const int s0 = hi * 2;                 // K offset within group-of-4 (0 or 2)

    // ---- Q tile as sixteen 16x4 A-fragments (K = 0..63), loaded once ----
    const float* qrow = q + ((size_t)bh * S_ + (size_t)(q0 + m)) * D_ + s0;
    v2f qa[16];
#pragma unroll
    for (int kb = 0; kb < 16; ++kb) {
        float2 tv = *(const float2*)(qrow + 4 * kb);
        qa[kb].x = tv.x; qa[kb].y = tv.y;
    }

    const float* kbase = kmat + (size_t)bh * S_ * D_;
    const int*   mbase = mask + (size_t)bh * S_ * S_ + (size_t)q0 * S_;

    // ---- phase 1: raw masked scores -> LDS ----
    for (int jt = wave; jt < S_ / 16; jt += 8) {
        const float* krow = kbase + (size_t)(jt * 16 + m) * D_ + s0;
        v2f kf[16];
#pragma unroll
        for (int kb = 0; kb < 16; ++kb) {
            float2 tv = *(const float2*)(krow + 4 * kb);
            kf[kb].x = tv.x; kf[kb].y = tv.y;
        }
        v8f acc = {};
#pragma unroll
        for (int kb = 0; kb < 16; ++kb) {
            acc = __builtin_amdgcn_wmma_f32_16x16x4_f32(
                false, qa[kb], false, kf[kb], (short)0, acc, false, false);
        }
        const int col = jt * 16 + m;
#pragma unroll
        for (int r = 0; r < 8; ++r) {
            int   row = r + 8 * hi;                       // C layout: VGPR r -> M=r / M=r+8
            int   mv  = __builtin_nontemporal_load(&mbase[(size_t)row * S_ + col]);
            float sv  = (mv != 0) ? acc[r] * 0.125f : -1.0e6f;
            sc[row * SROW + col] = sv;
        }
    }
    __syncthreads();

    // ---- phase 2: softmax over each 2048-wide row ----
    const int row2 = t >> 4;                // 16 threads per row
    const int sub  = t & 15;

    float mx = -INFINITY;
    for (int i = 0; i < 128; ++i)
        mx = fmaxf(mx, sc[row2 * SROW + sub + 16 * i]);
    pm[row2][sub] = mx;
    __syncthreads();
    if (t < 16) {
        float vv = pm[t][0];
        for (int i = 1; i < 16; ++i) vv = fmaxf(vv, pm[t][i]);
        rowmax[t] = vv;
    }
    __syncthreads();

    const float rm = rowmax[row2];
    float sum = 0.0f;
    for (int i = 0; i < 128; ++i) {
        int   idx = row2 * SROW + sub + 16 * i;
        float e   = expf(sc[idx] - rm);
        sc[idx]   = e;
        sum      += e;
    }
    ps[row2][sub] = sum;
    __syncthreads();
    if (t < 16) {
        float vv = 0.0f;
        for (int i = 0; i < 16; ++i) vv += ps[t][i];
        rowinv[t] = 1.0f / vv;
    }
    __syncthreads();

    // ---- phase 3: normalized scores -> global, nontemporal b128 stores ----
    float* obase = out + (size_t)bh * S_ * S_ + (size_t)q0 * S_;
    for (int i = 0; i < 32; ++i) {
        int idx4 = i * 256 + t;             // 8192 float4 per block
        int row  = idx4 >> 9;               // 512 float4 per row
        int c4   = idx4 & 511;
        v4f val  = *(const v4f*)(&sc[row * SROW + 4 * c4]);
        float inv = rowinv[row];
        val *= inv;
        __builtin_nontemporal_store(val, (v4f*)(obase + (size_t)row * S_ + 4 * c4));
    }
}

extern "C" void kernel_launch(void* const* d_in, const int* in_sizes, int n_in,
                              void* d_out, int out_size, void* d_ws, size_t ws_size,
                              hipStream_t stream) {
    const float* q    = (const float*)d_in[0];
    const float* k    = (const float*)d_in[1];
    const float* v    = (const float*)d_in[2];
    const int*   mask = (const int*)d_in[3];
    float*       out  = (float*)d_out;

    const size_t score_elems = (size_t)B_ * H_ * S_ * S_;   // 134,217,728
    const dim3 grid((B_ * H_) * (S_ / 16));                 // 4096 blocks
    const dim3 block(256);
    const size_t ldsBytes = (size_t)16 * SROW * sizeof(float);  // ~128 KB

    attn_score_kernel<<<grid, block, ldsBytes, stream>>>(q, k, mask, out);

    // second tuple element: verbatim copy of v
    hipMemcpyAsync(out + score_elems, v,
                   (size_t)in_sizes[2] * sizeof(float),
                   hipMemcpyDeviceToDevice, stream);
}